// DiffJPEGCoding_45775761440810
// MI455X (gfx1250) — compile-verified
//
#include <hip/hip_runtime.h>
#include <math.h>

typedef float v2f __attribute__((ext_vector_type(2)));
typedef float v8f __attribute__((ext_vector_type(8)));

#define WAVES 8
#define WPITCH 17
#define YPITCH 34

// ---------- DCT basis, exact float32 of cos((2x+1)u*pi/16) ----------
#define K1 0.98078528040323044913f
#define K2 0.92387953251128675613f
#define K3 0.83146961230254523708f
#define K4 0.70710678118654752440f
#define K5 0.55557023301960222474f
#define K6 0.38268343236508977173f
#define K7 0.19509032201612826785f

__device__ __constant__ float C8TAB[64] = {
  1.f,  K1,  K2,  K3,  K4,  K5,  K6,  K7,
  1.f,  K3,  K6, -K7, -K4, -K1, -K2, -K5,
  1.f,  K5, -K6, -K1, -K4,  K7,  K2,  K3,
  1.f,  K7, -K2, -K5,  K4,  K3, -K6, -K1,
  1.f, -K7, -K2,  K5,  K4, -K3, -K6,  K1,
  1.f, -K5, -K6,  K1, -K4, -K7,  K2, -K3,
  1.f, -K3,  K6,  K7, -K4,  K1, -K2,  K5,
  1.f, -K1,  K2, -K3,  K4, -K5,  K6, -K7,
};

// G = blockdiag(C, C), 16x16
__device__ __forceinline__ float gmat(int i, int j) {
  return ((i >> 3) == (j >> 3)) ? C8TAB[(i & 7) * 8 + (j & 7)] : 0.0f;
}

// ---------- reference math helpers ----------
__device__ __forceinline__ float round_poly(float x) {
  float r = rintf(x);            // round half-to-even, matches jnp.round
  float d = x - r;
  return r + d * d * d;
}

__device__ __forceinline__ float clip_soft(float x, float lo, float hi) {
  const float s = 0.02f;
  float outv = x;
  if (x > hi) outv = -s * (__expf(fminf(hi - x, 0.0f)) - 1.0f) + hi;
  if (x < lo) outv =  s * (__expf(fminf(x - lo, 0.0f)) - 1.0f) + lo;
  return outv;
}

// Wave-local LDS ordering fence: per-wave LDS ops are processed in order
// (DScnt, in order); this blocks compiler reordering and drains DS before reuse.
__device__ __forceinline__ void lds_fence() {
  asm volatile("s_wait_dscnt 0x0" ::: "memory");
}

// ---------- WMMA: 16x16x16 fp32 matmul via 4 chained 16x16x4 WMMAs ----------
__device__ __forceinline__ v8f wmma_k16(const float* a8, const float* b8, v8f acc) {
#pragma unroll
  for (int k = 0; k < 4; ++k) {
    v2f a = { a8[2 * k], a8[2 * k + 1] };
    v2f b = { b8[2 * k], b8[2 * k + 1] };
    acc = __builtin_amdgcn_wmma_f32_16x16x4_f32(false, a, false, b, (short)0, acc,
                                                false, false);
  }
  return acc;
}

// B-operand gather from LDS (chunk k / reg v on lane l: r = 4k+v+2*(l>>4), c = l&15)
__device__ __forceinline__ void read_b_t(const float* lds, int lane, float* bv) {
  const int n = lane & 15, hi = lane >> 4;
#pragma unroll
  for (int k = 0; k < 4; ++k)
#pragma unroll
    for (int v = 0; v < 2; ++v)
      bv[2 * k + v] = lds[n * WPITCH + (4 * k + v + 2 * hi)];   // transposed read
}

// ---------- DCT -> quant(round_poly) -> dequant -> IDCT on one 16x16 macro tile
//   T   = G^T X            (mm1)
//   Y^T = G^T T^T          (mm2)  elementwise quant on Y^T (qt transposed idx)
//   U   = G  F             (mm3)
//   Z^T = G  U^T           (mm4)  store un-transposed back to tile
__device__ __forceinline__ void jpeg_macro(float* tile, int pitch, float center,
                                           float* wb, const float* qt,
                                           const float* rqt,
                                           const float* aT, const float* aG,
                                           int lane) {
  const int n = lane & 15, hi = lane >> 4;
  float bv[8];

  // mm1: B = X (input, centered)
#pragma unroll
  for (int k = 0; k < 4; ++k)
#pragma unroll
    for (int v = 0; v < 2; ++v)
      bv[2 * k + v] = tile[(4 * k + v + 2 * hi) * pitch + n] - center;
  v8f acc = {0.f, 0.f, 0.f, 0.f, 0.f, 0.f, 0.f, 0.f};
  acc = wmma_k16(aT, bv, acc);

  lds_fence();                               // protect wb reuse across macros
#pragma unroll
  for (int j = 0; j < 8; ++j) wb[(j + 8 * hi) * WPITCH + n] = acc[j];   // T natural
  lds_fence();

  // mm2: B = T^T  ->  acc holds Yraw^T in D-layout
  read_b_t(wb, lane, bv);
  v8f acc2 = {0.f, 0.f, 0.f, 0.f, 0.f, 0.f, 0.f, 0.f};
  acc2 = wmma_k16(aT, bv, acc2);

  // elementwise quantization (transposed): element (m=j+8hi, n) = Yraw[n][m]
  // frequency u = n%8 = lane&7, v = j; alpha outer product is symmetric.
  const float rs2 = 0.70710678118654752f;
  const float au = ((lane & 7) == 0) ? rs2 : 1.0f;
#pragma unroll
  for (int j = 0; j < 8; ++j) {
    float av = (j == 0) ? rs2 : 1.0f;
    float af = 0.25f * au * av;
    float e  = round_poly(af * acc2[j] * rqt[(lane & 7) * 8 + j]);
    acc2[j]  = af * e * qt[(lane & 7) * 8 + j];          // F^T element
  }

#pragma unroll
  for (int j = 0; j < 8; ++j) wb[(j + 8 * hi) * WPITCH + n] = acc2[j];  // F^T natural
  lds_fence();

  // mm3: B = F (transposed read of stored F^T) -> U natural in D-layout
  read_b_t(wb, lane, bv);
  v8f acc3 = {0.f, 0.f, 0.f, 0.f, 0.f, 0.f, 0.f, 0.f};
  acc3 = wmma_k16(aG, bv, acc3);

  lds_fence();
#pragma unroll
  for (int j = 0; j < 8; ++j) wb[(j + 8 * hi) * WPITCH + n] = acc3[j];  // U natural
  lds_fence();

  // mm4: B = U^T -> Z^T in D-layout
  read_b_t(wb, lane, bv);
  v8f acc4 = {0.f, 0.f, 0.f, 0.f, 0.f, 0.f, 0.f, 0.f};
  acc4 = wmma_k16(aG, bv, acc4);

  // element (m, n) = Z[n][m]: spatial row n, col m -> un-transpose on store
#pragma unroll
  for (int j = 0; j < 8; ++j) tile[n * pitch + (j + 8 * hi)] = acc4[j] + center;
  lds_fence();
}

// ---------- kernel 1: per-batch quant table scaling (+ reciprocals) ----------
__global__ void qt_scale_kernel(const float* __restrict__ qual,
                                const float* __restrict__ qty,
                                const float* __restrict__ qtc,
                                float* __restrict__ ws) {
  int b = blockIdx.x, i = threadIdx.x;
  float q = qual[b];
  float s = (q < 50.0f) ? (5000.0f / q) : (200.0f - 2.0f * q);
  s = round_poly(s - 0.5f);                                       // floor_poly
  float fy = clip_soft(round_poly((qty[i] * s + 50.0f) / 100.0f - 0.5f), 1.0f, 255.0f);
  float fc = clip_soft(round_poly((qtc[i] * s + 50.0f) / 100.0f - 0.5f), 1.0f, 255.0f);
  ws[b * 64 + i]        = fy;
  ws[2048 + b * 64 + i] = fc;
  ws[4096 + b * 64 + i] = 1.0f / fy;
  ws[6144 + b * 64 + i] = 1.0f / fc;
}

// ---------- kernel 2: fused JPEG round-trip; one wave32 = one 16x32 pixel tile
__global__ __launch_bounds__(256, 1)
void jpeg_main_kernel(const float* __restrict__ img,
                      const float* __restrict__ qtws,
                      float* __restrict__ out) {
  __shared__ float qts[256];                 // [qy | qc | rqy | rqc], 64 each
  __shared__ float ybuf[WAVES][16][YPITCH];  // luma 16x32 (+pad, float2-aligned)
  __shared__ float cbuf[WAVES][16][WPITCH];  // rows 0-7: Cb 8x16, rows 8-15: Cr 8x16
  __shared__ float wbuf[WAVES][16][WPITCH];  // matmul staging

  const int tid  = threadIdx.x;
  const int wave = tid >> 5, lane = tid & 31;
  const int half = lane >> 4;                // 0: even row, 1: odd row of a pair
  const int cp   = lane & 15;                // column-pair index (2 cols each)
  const int b    = blockIdx.x >> 6;          // 64 blocks per image
  const int tile = (blockIdx.x & 63) * WAVES + wave;   // 0..511
  const int ty   = tile >> 4;                // 32 row-tiles of 16 rows
  const int tx   = tile & 15;                // 16 col-tiles of 32 cols
  const int row0 = ty * 16, col0 = tx * 32;

  qts[tid] = qtws[(tid >> 6) * 2048 + b * 64 + (tid & 63)];
  __syncthreads();                           // only cross-wave dependency

  // ---- stage 1: float2 loads, color convert, 2x2 chroma pool via shfl
  const size_t plane = (size_t)512 * 512;
  const float* rP = img + ((size_t)b * 3 + 0) * plane;
  const float* gP = img + ((size_t)b * 3 + 1) * plane;
  const float* bP = img + ((size_t)b * 3 + 2) * plane;
#pragma unroll
  for (int rp = 0; rp < 8; ++rp) {
    const int rloc = rp * 2 + half;
    const size_t off = (size_t)(row0 + rloc) * 512 + col0 + 2 * cp;
    float2 R  = *(const float2*)(rP + off);
    float2 G  = *(const float2*)(gP + off);
    float2 Bv = *(const float2*)(bP + off);
    float y0  =  0.299f    * R.x + 0.587f    * G.x + 0.114f    * Bv.x;
    float y1  =  0.299f    * R.y + 0.587f    * G.y + 0.114f    * Bv.y;
    float cbs = -0.168736f * (R.x + R.y) - 0.331264f * (G.x + G.y) + 0.5f * (Bv.x + Bv.y);
    float crs =  0.5f      * (R.x + R.y) - 0.418688f * (G.x + G.y) - 0.081312f * (Bv.x + Bv.y);
    *(float2*)&ybuf[wave][rloc][2 * cp] = make_float2(y0, y1);
    float cb4 = cbs + __shfl_xor(cbs, 16);   // add the other row of the pair
    float cr4 = crs + __shfl_xor(crs, 16);
    if (half == 0) {                          // centered chroma (+128/-128 cancel)
      cbuf[wave][rp][cp]     = 0.25f * cb4;
      cbuf[wave][8 + rp][cp] = 0.25f * cr4;
    }
  }
  lds_fence();

  // ---- stage 2: per-lane constant A-operands for G^T (DCT) and G (IDCT)
  float aT[8], aG[8];
  {
    const int m = lane & 15, hh = lane >> 4;
#pragma unroll
    for (int k = 0; k < 4; ++k)
#pragma unroll
      for (int v = 0; v < 2; ++v) {
        int K = 4 * k + v + 2 * hh;
        aT[2 * k + v] = gmat(K, m);   // (G^T)[m][K]
        aG[2 * k + v] = gmat(m, K);   // G[m][K]
      }
  }

  // ---- stage 3: three 16x16 macro tiles through DCT/quant/IDCT (WMMA)
  jpeg_macro(&ybuf[wave][0][0] + 0,  YPITCH, 128.0f, &wbuf[wave][0][0],
             &qts[0],  &qts[128], aT, aG, lane);
  jpeg_macro(&ybuf[wave][0][0] + 16, YPITCH, 128.0f, &wbuf[wave][0][0],
             &qts[0],  &qts[128], aT, aG, lane);
  jpeg_macro(&cbuf[wave][0][0],      WPITCH, 0.0f,   &wbuf[wave][0][0],
             &qts[64], &qts[192], aT, aG, lane);

  // ---- stage 4: upsample chroma, YCbCr->RGB, clip_soft, float2 stores
  float* oR = out + ((size_t)b * 3 + 0) * plane;
  float* oG = out + ((size_t)b * 3 + 1) * plane;
  float* oB = out + ((size_t)b * 3 + 2) * plane;
#pragma unroll
  for (int rp = 0; rp < 8; ++rp) {
    const int rloc = rp * 2 + half;
    float2 Y2 = *(float2*)&ybuf[wave][rloc][2 * cp];     // includes +128 restore
    float Cbc = cbuf[wave][rp][cp];                      // shared by 2x2 pixels
    float Crc = cbuf[wave][8 + rp][cp];
    float rr = 1.402f * Crc;
    float gg = -0.344136f * Cbc - 0.714136f * Crc;
    float bb = 1.772f * Cbc;
    const size_t off = (size_t)(row0 + rloc) * 512 + col0 + 2 * cp;
    *(float2*)(oR + off) = make_float2(clip_soft(Y2.x + rr, 0.0f, 255.0f),
                                       clip_soft(Y2.y + rr, 0.0f, 255.0f));
    *(float2*)(oG + off) = make_float2(clip_soft(Y2.x + gg, 0.0f, 255.0f),
                                       clip_soft(Y2.y + gg, 0.0f, 255.0f));
    *(float2*)(oB + off) = make_float2(clip_soft(Y2.x + bb, 0.0f, 255.0f),
                                       clip_soft(Y2.y + bb, 0.0f, 255.0f));
  }
}

extern "C" void kernel_launch(void* const* d_in, const int* in_sizes, int n_in,
                              void* d_out, int out_size, void* d_ws, size_t ws_size,
                              hipStream_t stream) {
  (void)n_in; (void)out_size; (void)ws_size;
  const float* img  = (const float*)d_in[0];   // (32,3,512,512)
  const float* qual = (const float*)d_in[1];   // (32,)
  const float* qty  = (const float*)d_in[2];   // (8,8)
  const float* qtc  = (const float*)d_in[3];   // (8,8)
  float* out = (float*)d_out;
  float* ws  = (float*)d_ws;                   // needs 8192 floats (32 KB)

  const int B = in_sizes[1];                   // 32
  qt_scale_kernel<<<B, 64, 0, stream>>>(qual, qty, qtc, ws);
  // 64 blocks per image * 8 waves per block * (16x32) tiles covers 512x512
  jpeg_main_kernel<<<B * 64, 256, 0, stream>>>(img, ws, out);
}